// trellis_quantizer_86569360818549
// MI455X (gfx1250) — compile-verified
//
#include <hip/hip_runtime.h>
#include <stdint.h>

// Trellis constants from the reference: L=16, K=2, V=1.
#define KV       2
#define S_RED    16384          // reduced states
#define NG       4096           // groups of 4 consecutive reduced states
#define BB       32             // batch
#define TT       64             // time steps (V=1 -> Tv = T)
#define NTHREADS 1024           // 32 wave32 waves on one WGP

#if defined(__gfx1250__) && __has_builtin(__builtin_amdgcn_global_load_async_to_lds_b32)
#define HAVE_ASYNC_LDS 1
#else
#define HAVE_ASYNC_LDS 0
#endif

#if HAVE_ASYNC_LDS
typedef __attribute__((address_space(1))) int gas_int;  // global
typedef __attribute__((address_space(3))) int las_int;  // LDS
#endif

// One workgroup per batch, one WGP per workgroup.
// LDS budget (of the WGP's 320KB):
//   cost[16384] f32 (in-place updated)      =  64 KB
//   codes[63][1024] u32 survivor words      = 252 KB
//   xs[64] + swv[32] + swi[32]              = ~0.5 KB
// total ~316.5 KB. Survivors for t=1..63 never touch global memory and the
// backtrack pointer-chase runs through LDS instead of L2. Only the t=0 codes
// (4KB/batch) go to the global workspace (one dependent load at chase end).
__global__ __launch_bounds__(NTHREADS, 1)
void viterbi_fwd_bwd(const float* __restrict__ X,
                     const float* __restrict__ lut,
                     float* __restrict__ out,
                     uint8_t* __restrict__ bp0g /* t=0 codes: [BB][NG] bytes */)
{
    __shared__ float    cost[S_RED];
    __shared__ uint32_t codes[(TT - 1) * NTHREADS];  // word per (t,tid); byte i = group tid+i*1024
    __shared__ float    xs[TT];
    __shared__ float    swv[32];
    __shared__ int      swi[32];

    const int b   = blockIdx.x;
    const int tid = threadIdx.x;

    // ---- Stage this batch's observation row into LDS (gfx1250 async copy) ----
    if (tid < TT) {
#if HAVE_ASYNC_LDS
        __builtin_amdgcn_global_load_async_to_lds_b32(
            (gas_int*)(X + b * TT + tid),
            (las_int*)&xs[tid],
            0, 0);
#else
        xs[tid] = X[b * TT + tid];
#endif
    }
    // Warm the 256KB decode LUT into cache (global_prefetch_b8).
    __builtin_prefetch(lut + (tid << 6), 0, 1);

#if HAVE_ASYNC_LDS
#if __has_builtin(__builtin_amdgcn_s_wait_asynccnt)
    __builtin_amdgcn_s_wait_asynccnt(0);
#else
    asm volatile("s_wait_asynccnt 0" ::: "memory");
#endif
#endif
    __syncthreads();

    // ---- t = 0: cost[r] = min_j (lut[4r+j]-x0)^2 ; code0[r] = argmin_j ----
    {
        const float x0 = xs[0];
        uint8_t* bp0 = bp0g + (size_t)b * NG;
#pragma unroll
        for (int i = 0; i < 4; ++i) {
            const int g = tid + (i << 10);          // group of 4 reduced states
            uint32_t packed = 0;
            float nc[4];
#pragma unroll
            for (int j = 0; j < 4; ++j) {
                const int r = (g << 2) + j;
                float bv; int bd = 0;
#pragma unroll
                for (int d = 0; d < 4; ++d) {
                    const float df = lut[(r << 2) + d] - x0;
                    const float e  = df * df;
                    if (d == 0)      { bv = e; }
                    else if (e < bv) { bv = e; bd = d; }
                }
                nc[j] = bv;
                packed |= (uint32_t)bd << (2 * j);
            }
            *reinterpret_cast<float4*>(&cost[g << 2]) =
                make_float4(nc[0], nc[1], nc[2], nc[3]);
            bp0[g] = (uint8_t)packed;
        }
    }
    __syncthreads();

    // ---- Forward recurrence, single in-place cost buffer ----
    // Phase 1: gather the 16 predecessor costs this thread needs into VGPRs.
    // Phase 2 (after barrier): recompute + overwrite cost in place, pack survivors.
    for (int t = 1; t < TT; ++t) {
        const float x = xs[t];
        float c[4][4];
#pragma unroll
        for (int i = 0; i < 4; ++i) {
            const int g = tid + (i << 10);
#pragma unroll
            for (int d = 0; d < 4; ++d)
                c[i][d] = cost[g + (d << 12)];
        }
        __syncthreads();

        uint32_t word = 0;
#pragma unroll
        for (int i = 0; i < 4; ++i) {
            const int g = tid + (i << 10);
            uint32_t packed = 0;
            float nc[4];
#pragma unroll
            for (int j = 0; j < 4; ++j) {
                const int s = (g << 2) + j;          // reduced state index
                const float d0 = lut[s]         - x;
                const float d1 = lut[s + 16384] - x;
                const float d2 = lut[s + 32768] - x;
                const float d3 = lut[s + 49152] - x;
                const float t0 = fmaf(d0, d0, c[i][0]);
                const float t1 = fmaf(d1, d1, c[i][1]);
                const float t2 = fmaf(d2, d2, c[i][2]);
                const float t3 = fmaf(d3, d3, c[i][3]);
                float bv = t0; int bd = 0;           // strict < -> first-occurrence argmin
                if (t1 < bv) { bv = t1; bd = 1; }
                if (t2 < bv) { bv = t2; bd = 2; }
                if (t3 < bv) { bv = t3; bd = 3; }
                nc[j] = bv;
                packed |= (uint32_t)bd << (2 * j);
            }
            *reinterpret_cast<float4*>(&cost[g << 2]) =
                make_float4(nc[0], nc[1], nc[2], nc[3]);
            word |= packed << (8 * i);
        }
        codes[((t - 1) << 10) + tid] = word;         // one conflict-free ds_store_b32
        __syncthreads();
    }

    // Release the t=0 survivor bytes in global memory for the backtracker.
    __threadfence();

    // ---- argmin over 16384 final costs (first-occurrence on ties) ----
    float bestv = INFINITY;
    int   besti = 0x7fffffff;
#pragma unroll
    for (int i = 0; i < 4; ++i) {
        const int g = tid + (i << 10);
#pragma unroll
        for (int j = 0; j < 4; ++j) {
            const int r = (g << 2) + j;
            const float v = cost[r];
            if (v < bestv || (v == bestv && r < besti)) { bestv = v; besti = r; }
        }
    }
#pragma unroll
    for (int m = 16; m >= 1; m >>= 1) {              // wave32 butterfly reduce
        const float ov = __shfl_xor(bestv, m, 32);
        const int   oi = __shfl_xor(besti, m, 32);
        if (ov < bestv || (ov == bestv && oi < besti)) { bestv = ov; besti = oi; }
    }
    if ((tid & 31) == 0) { swv[tid >> 5] = bestv; swi[tid >> 5] = besti; }
    __syncthreads();

    if (tid < 32) {                                  // wave 0 reduces the 32 wave results
        bestv = swv[tid];
        besti = swi[tid];
#pragma unroll
        for (int m = 16; m >= 1; m >>= 1) {
            const float ov = __shfl_xor(bestv, m, 32);
            const int   oi = __shfl_xor(besti, m, 32);
            if (ov < bestv || (ov == bestv && oi < besti)) { bestv = ov; besti = oi; }
        }

        // ---- Backtrack (LDS pointer-chase) + emit hatX and final_state ----
        if (tid == 0) {
            int* st_out = reinterpret_cast<int*>(out) + BB * TT;

            const int rstar = besti;
            int g = rstar >> 2;
            uint32_t w  = codes[((TT - 2) << 10) + (g & 1023)];
            uint32_t cb = (w >> ((g >> 10) * 8)) & 0xffu;
            int code = (cb >> ((rstar & 3) * 2)) & 3;
            int s = rstar + (code << (16 - KV));     // from_state[-1]
            out[b * TT + (TT - 1)]    = lut[s];
            st_out[b * TT + (TT - 1)] = s;

            int nxt = s;
            for (int i2 = TT - 2; i2 >= 1; --i2) {
                const int r = nxt >> KV;
                g  = r >> 2;
                w  = codes[((i2 - 1) << 10) + (g & 1023)];
                cb = (w >> ((g >> 10) * 8)) & 0xffu;
                code = (cb >> ((r & 3) * 2)) & 3;
                s = r + (code << (16 - KV));         // prev_state at step i2
                out[b * TT + i2]    = lut[s];
                st_out[b * TT + i2] = s;
                nxt = s;
            }
            {
                const int r = nxt >> KV;
                const uint8_t cb0 = bp0g[(size_t)b * NG + (r >> 2)];
                code = (cb0 >> ((r & 3) * 2)) & 3;
                s = (r << KV) + code;                // full0
                out[b * TT + 0]    = lut[s];
                st_out[b * TT + 0] = s;
            }
        }
    }
}

extern "C" void kernel_launch(void* const* d_in, const int* in_sizes, int n_in,
                              void* d_out, int out_size, void* d_ws, size_t ws_size,
                              hipStream_t stream) {
    (void)in_sizes; (void)n_in; (void)out_size; (void)ws_size;
    const float* X   = (const float*)d_in[0];   // (32, 64) f32
    const float* lut = (const float*)d_in[1];   // (65536, 1) f32
    float* out = (float*)d_out;                 // [hatX (2048 f32) | final_state (2048 i32)]
    uint8_t* bp0g = (uint8_t*)d_ws;             // t=0 survivor codes: 32*4096 = 128KB scratch
    viterbi_fwd_bwd<<<BB, NTHREADS, 0, stream>>>(X, lut, out, bp0g);
}